// FNO1dOptimized_51376398795180
// MI455X (gfx1250) — compile-verified
//
#include <hip/hip_runtime.h>
#include <math.h>

// ---------------------------------------------------------------------------
// FNO-1d with continuous (Chebyshev-quadrature) Fourier transform, collapsed
// to rank-16 spectral operators + f32 WMMA GEMMs for MI455X (gfx1250, wave32).
// B=8, n=1024, W=64, MODES=16, L=100, M=20 (J=2000 quadrature nodes).
// ---------------------------------------------------------------------------

typedef __attribute__((ext_vector_type(2))) float v2f;
typedef __attribute__((ext_vector_type(8))) float v8f;

#define PI_F 3.14159265358979323846f

__device__ __forceinline__ float gelu_exact(float x) {
  // jax.nn.gelu(approximate=False) = x * 0.5 * (1 + erf(x/sqrt(2)))
  return 0.5f * x * (1.0f + erff(x * 0.7071067811865476f));
}

// --- Kernel 1: Chebyshev node tables for the t-domain (CFT) and f-domain
// (ICFT) quadratures: node position, weight, interp cell i0 and fraction.
__global__ void k_setup_nodes(float* tn_node, float* tn_w, float* tn_frac, int* tn_i0,
                              float* fn_node, float* fn_w, float* fn_frac, int* fn_i0) {
  int idx = blockIdx.x * blockDim.x + threadIdx.x;
  if (idx >= 4000) return;
  int which = idx / 2000;          // 0: t-nodes over [0,1], 1: f-nodes over [-512,511]
  int j = idx % 2000;
  int l = j / 20, m = j % 20;
  float u = cosf((2.0f * (float)m + 1.0f) * PI_F / 40.0f);
  float a = which ? -512.0f : 0.0f;
  float b = which ? 511.0f : 1.0f;
  float h = (b - a) * 0.01f;       // /L, L=100
  float node = (a + ((float)l + 0.5f) * h) + 0.5f * h * u;
  float w = 0.5f * h * (PI_F / 20.0f) * sqrtf(fmaxf(1.0f - u * u, 0.0f));
  float dx = which ? 1.0f : (1.0f / 1023.0f);
  float pos = (node - a) / dx;
  int i0 = (int)floorf(pos);
  i0 = max(0, min(i0, 1022));
  float frac = pos - (float)i0;
  if (which == 0) { tn_node[j] = node; tn_w[j] = w; tn_frac[j] = frac; tn_i0[j] = i0; }
  else            { fn_node[j] = node; fn_w[j] = w; fn_frac[j] = frac; fn_i0[j] = i0; }
}

// --- Kernel 2: G[n][k] (1024 x 16 complex): CFT collapsed onto the t-grid.
// Xc[b,c,k] = sum_n h[b,c,n] * G[n,k]
__global__ void k_compute_G(const float* __restrict__ tn_node, const float* __restrict__ tn_w,
                            const float* __restrict__ tn_frac, const int* __restrict__ tn_i0,
                            float2* __restrict__ G) {
  int n = blockIdx.x * blockDim.x + threadIdx.x;
  if (n >= 1024) return;
  float gr[16], gi[16];
#pragma unroll
  for (int k = 0; k < 16; ++k) { gr[k] = 0.0f; gi[k] = 0.0f; }
  for (int j = 0; j < 2000; ++j) {
    int i0 = tn_i0[j];
    float coef;
    if (i0 == n)          coef = 1.0f - tn_frac[j];   // v0 coefficient
    else if (i0 == n - 1) coef = tn_frac[j];          // v1 coefficient
    else continue;
    float cw = coef * tn_w[j];
    float node = tn_node[j];
#pragma unroll
    for (int k = 0; k < 16; ++k) {                    // f_k = k for k<16
      float ang = -2.0f * PI_F * (float)k * node;
      float s, c; __sincosf(ang, &s, &c);
      gr[k] += cw * c; gi[k] += cw * s;
    }
  }
#pragma unroll
  for (int k = 0; k < 16; ++k) G[n * 16 + k] = make_float2(gr[k], gi[k]);
}

// --- Kernel 3: Econ rows (32 x 1024): ICFT collapsed onto the 16 live modes.
// row k = Re part, row 16+k = Im part; x1 = ym_r@Er - ym_i@Ei.
__global__ void k_compute_E(const float* __restrict__ fn_node, const float* __restrict__ fn_w,
                            const float* __restrict__ fn_frac, const int* __restrict__ fn_i0,
                            float* __restrict__ Econ) {
  int tid = blockIdx.x * blockDim.x + threadIdx.x;
  if (tid >= 16 * 1024) return;
  int k = tid >> 10, t = tid & 1023;
  float tv = (float)t * (1.0f / 1023.0f);
  int target = 512 + k;             // sorted-frequency slot of mode k
  float er = 0.0f, ei = 0.0f;
  for (int j = 0; j < 2000; ++j) {
    int i0 = fn_i0[j];
    float coef;
    if (i0 == target)          coef = 1.0f - fn_frac[j];
    else if (i0 + 1 == target) coef = fn_frac[j];
    else continue;
    float cw = coef * fn_w[j];
    float ang = 2.0f * PI_F * fn_node[j] * tv;
    float s, c; __sincosf(ang, &s, &c);
    er += cw * c; ei += cw * s;
  }
  Econ[k * 1024 + t] = er;
  Econ[(16 + k) * 1024 + t] = ei;
}

// --- Kernel 4: fc0 lift: h[b,c,n] = fc0_w[c,0]*x[b,n] + fc0_w[c,1]*grid[n] + fc0_b[c]
__global__ void k_fc0(const float* __restrict__ x, const float* __restrict__ w,
                      const float* __restrict__ bvec, float* __restrict__ h) {
  int idx = blockIdx.x * blockDim.x + threadIdx.x;
  if (idx >= 8 * 64 * 1024) return;
  int n = idx & 1023;
  int c = (idx >> 10) & 63;
  int b = idx >> 16;
  float g = (float)n * (1.0f / 1023.0f);
  h[idx] = w[c * 2 + 0] * x[b * 1024 + n] + w[c * 2 + 1] * g + bvec[c];
}

// --- Kernel 5: CFT via precomputed G: Xc[b,i,k] = sum_n h[b,i,n]*G[n,k]
__global__ void k_cft(const float* __restrict__ h, const float2* __restrict__ G,
                      float2* __restrict__ Xc) {
  int tid = blockIdx.x * blockDim.x + threadIdx.x;
  if (tid >= 8 * 64 * 16) return;
  int k = tid & 15;
  int bi = tid >> 4;
  const float* hr = h + (size_t)bi * 1024;
  float ar = 0.0f, ai = 0.0f;
  for (int n = 0; n < 1024; ++n) {
    float hv = hr[n];
    float2 g = G[n * 16 + k];
    ar += hv * g.x; ai += hv * g.y;
  }
  Xc[tid] = make_float2(ar, ai);
}

// --- Kernel 6: per-mode complex channel mix: ym[b,o,k] = sum_i Xc[b,i,k]*spec[i,o,k]
__global__ void k_modemix(const float2* __restrict__ Xc, const float2* __restrict__ spec,
                          float2* __restrict__ ym) {
  int tid = blockIdx.x * blockDim.x + threadIdx.x;
  if (tid >= 8 * 64 * 16) return;
  int k = tid & 15;
  int o = (tid >> 4) & 63;
  int b = tid >> 10;
  float yr = 0.0f, yi = 0.0f;
  for (int i = 0; i < 64; ++i) {
    float2 xv = Xc[((b * 64 + i) << 4) + k];
    float2 s = spec[((i * 64 + o) << 4) + k];
    yr += xv.x * s.x - xv.y * s.y;
    yi += xv.x * s.y + xv.y * s.x;
  }
  ym[tid] = make_float2(yr, yi);
}

// --- Kernel 7: fused A = [w_w | ym_r | -ym_i] (8 x 64 x 96)
__global__ void k_buildA(const float* __restrict__ ww, const float2* __restrict__ ym,
                         float* __restrict__ Abuf) {
  int tid = blockIdx.x * blockDim.x + threadIdx.x;
  if (tid >= 8 * 64 * 96) return;
  int q = tid % 96;
  int o = (tid / 96) % 64;
  int b = tid / (96 * 64);
  float v;
  if (q < 64)      v = ww[o * 64 + q];
  else if (q < 80) v = ym[((b * 64 + o) << 4) + (q - 64)].x;
  else             v = -ym[((b * 64 + o) << 4) + (q - 80)].y;
  Abuf[tid] = v;
}

// --- WMMA f32 GEMM: D(M x 1024) = A(M x KTOT) * B(KTOT x 1024) + bias, opt GELU.
// B rows 0..63 come from the per-batch activation h[b]; rows 64..KTOT-1 from Econ.
// One wave32 per 16x16 tile; K stepped by 4 via V_WMMA_F32_16X16X4_F32.
template <int KTOT>
__global__ void k_gemm_wmma(const float* __restrict__ A, int aBatchStride,
                            const float* __restrict__ Bmain, const float* __restrict__ Econ,
                            const float* __restrict__ bias, float* __restrict__ out,
                            int M, int doGelu) {
  const int lane = threadIdx.x & 31;
  const int wav  = threadIdx.x >> 5;
  const int l15  = lane & 15;
  const int half = lane >> 4;                       // selects K pair / M+8 rows
  const int n     = (blockIdx.x * 4 + wav) * 16 + l15;
  const int mtile = blockIdx.y;
  const int b     = blockIdx.z;

  const float* Ab = A + (size_t)b * aBatchStride + (size_t)(mtile * 16 + l15) * KTOT;
  const float* Bb = Bmain + (size_t)b * 64 * 1024;

  v8f acc = {0.0f, 0.0f, 0.0f, 0.0f, 0.0f, 0.0f, 0.0f, 0.0f};
#pragma unroll
  for (int k0 = 0; k0 < KTOT; k0 += 4) {
    const int k = k0 + half * 2;                    // lanes 0-15: K=k0,k0+1; 16-31: k0+2,k0+3
    v2f a, bf;
    a.x = Ab[k];
    a.y = Ab[k + 1];
    if (k0 < 64) {                                  // compile-time per unrolled step
      bf.x = Bb[(size_t)k * 1024 + n];
      bf.y = Bb[(size_t)(k + 1) * 1024 + n];
    } else {
      bf.x = Econ[(size_t)(k - 64) * 1024 + n];
      bf.y = Econ[(size_t)(k - 63) * 1024 + n];
    }
    acc = __builtin_amdgcn_wmma_f32_16x16x4_f32(false, a, false, bf, (short)0, acc,
                                                false, false);
  }
#pragma unroll
  for (int v = 0; v < 8; ++v) {                     // C/D layout: row v + 8*half, col l15
    int m = mtile * 16 + half * 8 + v;
    float val = acc[v] + bias[m];
    if (doGelu) val = gelu_exact(val);
    out[((size_t)b * M + m) * 1024 + n] = val;
  }
}

// --- Kernel 9: fc2 projection to scalar output
__global__ void k_fc2(const float* __restrict__ u1, const float* __restrict__ w,
                      const float* __restrict__ bvec, float* __restrict__ out) {
  int tid = blockIdx.x * blockDim.x + threadIdx.x;
  if (tid >= 8 * 1024) return;
  int n = tid & 1023, b = tid >> 10;
  float s = 0.0f;
  for (int p = 0; p < 128; ++p) s += w[p] * u1[((size_t)(b * 128 + p)) * 1024 + n];
  out[tid] = s + bvec[0];
}

extern "C" void kernel_launch(void* const* d_in, const int* in_sizes, int n_in,
                              void* d_out, int out_size, void* d_ws, size_t ws_size,
                              hipStream_t stream) {
  (void)in_sizes; (void)n_in; (void)out_size; (void)ws_size;
  const float*  x     = (const float*)d_in[0];
  const float*  fc0_w = (const float*)d_in[1];
  const float*  fc0_b = (const float*)d_in[2];
  const float2* spec[4] = {(const float2*)d_in[3], (const float2*)d_in[4],
                           (const float2*)d_in[5], (const float2*)d_in[6]};
  const float* ww[4] = {(const float*)d_in[7], (const float*)d_in[9],
                        (const float*)d_in[11], (const float*)d_in[13]};
  const float* wb[4] = {(const float*)d_in[8], (const float*)d_in[10],
                        (const float*)d_in[12], (const float*)d_in[14]};
  const float* fc1_w = (const float*)d_in[15];
  const float* fc1_b = (const float*)d_in[16];
  const float* fc2_w = (const float*)d_in[17];
  const float* fc2_b = (const float*)d_in[18];
  float* out = (float*)d_out;

  // Workspace layout (float offsets)
  float* ws = (float*)d_ws;
  float* tn_node = ws;            float* tn_w = ws + 2000;
  float* tn_frac = ws + 4000;     int*   tn_i0 = (int*)(ws + 6000);
  float* fn_node = ws + 8000;     float* fn_w = ws + 10000;
  float* fn_frac = ws + 12000;    int*   fn_i0 = (int*)(ws + 14000);
  float2* G    = (float2*)(ws + 16000);          // 1024*16 complex
  float*  Econ = ws + 48768;                     // 32*1024
  float*  hA   = ws + 81536;                     // 8*64*1024
  float*  hB   = ws + 605824;                    // 8*64*1024
  float2* Xc   = (float2*)(ws + 1130112);        // 8*64*16 complex
  float2* ym   = (float2*)(ws + 1146496);        // 8*64*16 complex
  float*  Abuf = ws + 1162880;                   // 8*64*96
  float*  u1   = ws + 1212032;                   // 8*128*1024

  k_setup_nodes<<<16, 256, 0, stream>>>(tn_node, tn_w, tn_frac, tn_i0,
                                        fn_node, fn_w, fn_frac, fn_i0);
  k_compute_G<<<4, 256, 0, stream>>>(tn_node, tn_w, tn_frac, tn_i0, G);
  k_compute_E<<<64, 256, 0, stream>>>(fn_node, fn_w, fn_frac, fn_i0, Econ);
  k_fc0<<<2048, 256, 0, stream>>>(x, fc0_w, fc0_b, hA);

  float* hcur = hA;
  float* hnext = hB;
  for (int i = 0; i < 4; ++i) {
    k_cft<<<32, 256, 0, stream>>>(hcur, G, Xc);
    k_modemix<<<32, 256, 0, stream>>>(Xc, spec[i], ym);
    k_buildA<<<192, 256, 0, stream>>>(ww[i], ym, Abuf);
    dim3 gL(16, 4, 8);   // n-tiles/4, M/16, batch
    k_gemm_wmma<96><<<gL, 128, 0, stream>>>(Abuf, 64 * 96, hcur, Econ, wb[i],
                                            hnext, 64, (i < 3) ? 1 : 0);
    float* tmp = hcur; hcur = hnext; hnext = tmp;
  }

  dim3 gF(16, 8, 8);     // fc1: M=128, K=64 (Econ path never taken)
  k_gemm_wmma<64><<<gF, 128, 0, stream>>>(fc1_w, 0, hcur, Econ, fc1_b, u1, 128, 1);
  k_fc2<<<32, 256, 0, stream>>>(u1, fc2_w, fc2_b, out);
}